// CNN_25744033972549
// MI455X (gfx1250) — compile-verified
//
#include <hip/hip_runtime.h>
#include <hip/hip_bf16.h>

typedef __attribute__((ext_vector_type(2))) float v2f;
typedef __attribute__((ext_vector_type(8))) float v8f;

#define BB    8
#define TT    4096
#define CIN   32
#define HH    256
#define CC2   512
#define OO    24

#define XTP   4112   // xt padded row: 9 left pad + 4096 + zero tail
#define H1P   4128   // h1 padded row: 19 left pad + 4096 + tail (never read)
#define HPP   4112   // hp padded row: 9 replicated head + 4096 + zero tail

#define K1C   12     // conv1 k padded 10 -> 12
#define K1ROW (CIN * K1C)      // 384  (wp1 row stride)
#define K2ROW 5120             // w2 row stride (256*20, used directly)
#define K3C   12     // conv3 k padded 10 -> 12
#define K3ROW (CC2 * K3C)      // 6144 (wp3 row stride)

__device__ __forceinline__ v8f wmma_f32(v2f a, v2f b, v8f c) {
  return __builtin_amdgcn_wmma_f32_16x16x4_f32(false, a, false, b, (short)0, c,
                                               false, false);
}

// ---------------------------------------------------------------------------
// Prologue kernels (tiny; all branchiness lives here, not in the GEMMs)
// ---------------------------------------------------------------------------

// wp1[m][ci*12 + k] = k<10 ? w1[m][ci][k] : 0        (256 x 384)
__global__ __launch_bounds__(256) void pack_w1_k(const float* __restrict__ w1,
                                                 float* __restrict__ wp1) {
  int i = blockIdx.x * 256 + threadIdx.x;            // 256*384 = 98304
  int m = i / K1ROW, r = i % K1ROW;
  int ci = r / K1C, k = r % K1C;
  wp1[i] = (k < 10) ? w1[m * 320 + ci * 10 + k] : 0.f;
}

// wp3[m][c*12 + k] = (m<24 && k<10) ? wf[m][k*512 + c] : 0    (32 x 6144)
__global__ __launch_bounds__(256) void pack_w3_k(const float* __restrict__ wf,
                                                 float* __restrict__ wp3) {
  int i = blockIdx.x * 256 + threadIdx.x;            // 32*6144 = 196608
  int m = i / K3ROW, r = i % K3ROW;
  int c = r / K3C, k = r % K3C;
  wp3[i] = (m < OO && k < 10) ? wf[m * 5120 + k * CC2 + c] : 0.f;
}

// xt[b][ci][j] = (9 <= j < 9+4096) ? x[b][j-9][ci] : 0        (8 x 32 x 4112)
__global__ __launch_bounds__(256) void transpose_pad_k(const float* __restrict__ x,
                                                       float* __restrict__ xt) {
  int i = blockIdx.x * 256 + threadIdx.x;            // 8*32*4112 = 1052672
  int j  = i % XTP;
  int ci = (i / XTP) % CIN;
  int b  = i / (XTP * CIN);
  int t  = j - 9;
  xt[i] = (t >= 0 && t < TT) ? x[((size_t)(b * TT + t)) * CIN + ci] : 0.f;
}

// zero the 19-sample left halo of every h1p row                (8*256*19)
__global__ __launch_bounds__(256) void zero_h1_halo_k(float* __restrict__ h1p) {
  int i = blockIdx.x * 256 + threadIdx.x;            // 38912 = 152*256
  int row = i / 19, off = i % 19;
  h1p[(size_t)row * H1P + off] = 0.f;
}

// hp head replicate (repeat(first,9)) + zero tail              (8*512*16)
__global__ __launch_bounds__(256) void fix_hp_k(float* __restrict__ hp) {
  int i = blockIdx.x * 256 + threadIdx.x;            // 65536 = 256*256
  int row = i >> 4, slot = i & 15;
  size_t base = (size_t)row * HPP;
  if (slot < 9) hp[base + slot] = hp[base + 9];
  else          hp[base + 4105 + (slot - 9)] = 0.f;
}

// ---------------------------------------------------------------------------
// Conv GEMMs: one wave computes 32(M) x 64(N), fp32 WMMA 16x16x4.
// Fragment mapping (wave32):
//   A 16x4 : lane<16 -> K pair {0,1}; lane>=16 -> K pair {2,3}; M = lane&15
//   B 4x16 : mirrored; N = lane&15
//   C/D    : VGPR e -> M = e + (lane>=16 ? 8 : 0); N = lane&15
// ---------------------------------------------------------------------------

// conv1: h1p[b][m][t+19] = b1[m] + sum_{ci,k} xt[b][ci][t+k] * wp1[m][ci*12+k]
__global__ __launch_bounds__(256) void conv1_k(const float* __restrict__ xt,
                                               const float* __restrict__ wp1,
                                               const float* __restrict__ b1,
                                               float* __restrict__ h1p) {
  const int lane = threadIdx.x & 31;
  const int unit = blockIdx.x * 8 + (threadIdx.x >> 5);   // 8mb * 8b * 64nb = 4096
  const int mblk  = unit >> 9;              // / (8*64)
  const int b     = (unit >> 6) & 7;
  const int nbase = (unit & 63) * 64;
  const int mbase = mblk * 32;
  const int mlo  = lane & 15;
  const int sel2 = (lane >> 4) << 1;
  const int mhi8 = (lane >> 4) << 3;
  const int t0   = nbase + mlo;

  v8f acc[2][4] = {};
  const float* xb  = xt + (size_t)b * CIN * XTP;
  const float* wr0 = wp1 + (size_t)(mbase + mlo) * K1ROW + sel2;
  const float* wr1 = wr0 + (size_t)16 * K1ROW;

  for (int ci = 0; ci < CIN; ++ci) {
    const float* px = xb + (size_t)ci * XTP + t0 + sel2;
    const int wo = ci * K1C;
#pragma unroll
    for (int k = 0; k < K1C; k += 4) {
      v2f a0 = *(const v2f*)(wr0 + wo + k);       // 8B aligned
      v2f a1 = *(const v2f*)(wr1 + wo + k);
      v2f f[4];
#pragma unroll
      for (int j = 0; j < 4; ++j) {
        f[j][0] = px[j * 16 + k];
        f[j][1] = px[j * 16 + k + 1];
      }
#pragma unroll
      for (int j = 0; j < 4; ++j) {
        acc[0][j] = wmma_f32(a0, f[j], acc[0][j]);
        acc[1][j] = wmma_f32(a1, f[j], acc[1][j]);
      }
    }
  }
#pragma unroll
  for (int i = 0; i < 2; ++i)
    for (int j = 0; j < 4; ++j) {
      const int n = nbase + j * 16 + mlo;
#pragma unroll
      for (int e = 0; e < 8; ++e) {
        const int m = mbase + i * 16 + mhi8 + e;
        h1p[((size_t)(b * HH + m)) * H1P + n + 19] = acc[i][j][e] + b1[m];
      }
    }
}

// conv2: hp[b][m][t+9] = b2[m] + sum_{ci,k} h1p[b][ci][t+k] * w2[m][ci*20+k]
__global__ __launch_bounds__(256) void conv2_k(const float* __restrict__ h1p,
                                               const float* __restrict__ w2,
                                               const float* __restrict__ b2,
                                               float* __restrict__ hp) {
  const int lane = threadIdx.x & 31;
  const int unit = blockIdx.x * 8 + (threadIdx.x >> 5);   // 16mb * 8b * 64nb = 8192
  const int mblk  = unit >> 9;
  const int b     = (unit >> 6) & 7;
  const int nbase = (unit & 63) * 64;
  const int mbase = mblk * 32;
  const int mlo  = lane & 15;
  const int sel2 = (lane >> 4) << 1;
  const int mhi8 = (lane >> 4) << 3;
  const int t0   = nbase + mlo;

  v8f acc[2][4] = {};
  const float* inb = h1p + (size_t)b * HH * H1P;
  const float* wr0 = w2 + (size_t)(mbase + mlo) * K2ROW + sel2;
  const float* wr1 = wr0 + (size_t)16 * K2ROW;

  for (int ci = 0; ci < HH; ++ci) {
    const float* px = inb + (size_t)ci * H1P + t0 + sel2;
    const int wo = ci * 20;
#pragma unroll
    for (int k = 0; k < 20; k += 4) {
      v2f a0 = *(const v2f*)(wr0 + wo + k);
      v2f a1 = *(const v2f*)(wr1 + wo + k);
      v2f f[4];
#pragma unroll
      for (int j = 0; j < 4; ++j) {
        f[j][0] = px[j * 16 + k];
        f[j][1] = px[j * 16 + k + 1];
      }
#pragma unroll
      for (int j = 0; j < 4; ++j) {
        acc[0][j] = wmma_f32(a0, f[j], acc[0][j]);
        acc[1][j] = wmma_f32(a1, f[j], acc[1][j]);
      }
    }
  }
#pragma unroll
  for (int i = 0; i < 2; ++i)
    for (int j = 0; j < 4; ++j) {
      const int n = nbase + j * 16 + mlo;
#pragma unroll
      for (int e = 0; e < 8; ++e) {
        const int m = mbase + i * 16 + mhi8 + e;
        hp[((size_t)(b * CC2 + m)) * HPP + n + 9] = acc[i][j][e] + b2[m];
      }
    }
}

// conv3: out[b][t][o] = bf[o] + sum_{c,k} hp[b][c][t+k] * wp3[o][c*12+k]
__global__ __launch_bounds__(256) void conv3_k(const float* __restrict__ hp,
                                               const float* __restrict__ wp3,
                                               const float* __restrict__ bf,
                                               float* __restrict__ out) {
  const int lane = threadIdx.x & 31;
  const int unit = blockIdx.x * 8 + (threadIdx.x >> 5);   // 8b * 64nb = 512
  const int b     = unit >> 6;
  const int nbase = (unit & 63) * 64;
  const int mlo  = lane & 15;
  const int sel2 = (lane >> 4) << 1;
  const int mhi8 = (lane >> 4) << 3;
  const int t0   = nbase + mlo;

  v8f acc[2][4] = {};
  const float* inb = hp + (size_t)b * CC2 * HPP;
  const float* wr0 = wp3 + (size_t)mlo * K3ROW + sel2;          // rows 0..15
  const float* wr1 = wr0 + (size_t)16 * K3ROW;                  // rows 16..31 (zero-padded)

  for (int c = 0; c < CC2; ++c) {
    const float* px = inb + (size_t)c * HPP + t0 + sel2;
    const int wo = c * K3C;
#pragma unroll
    for (int k = 0; k < K3C; k += 4) {
      v2f a0 = *(const v2f*)(wr0 + wo + k);
      v2f a1 = *(const v2f*)(wr1 + wo + k);
      v2f f[4];
#pragma unroll
      for (int j = 0; j < 4; ++j) {
        f[j][0] = px[j * 16 + k];
        f[j][1] = px[j * 16 + k + 1];
      }
#pragma unroll
      for (int j = 0; j < 4; ++j) {
        acc[0][j] = wmma_f32(a0, f[j], acc[0][j]);
        acc[1][j] = wmma_f32(a1, f[j], acc[1][j]);
      }
    }
  }
#pragma unroll
  for (int i = 0; i < 2; ++i)
    for (int j = 0; j < 4; ++j) {
      const int t = nbase + j * 16 + mlo;
#pragma unroll
      for (int e = 0; e < 8; ++e) {
        const int m = i * 16 + mhi8 + e;
        if (m < OO)
          out[((size_t)(b * TT + t)) * OO + m] = acc[i][j][e] + bf[m];
      }
    }
}

// ---------------------------------------------------------------------------
extern "C" void kernel_launch(void* const* d_in, const int* in_sizes, int n_in,
                              void* d_out, int out_size, void* d_ws, size_t ws_size,
                              hipStream_t stream) {
  const float* x  = (const float*)d_in[0];
  const float* w1 = (const float*)d_in[1];
  const float* b1 = (const float*)d_in[2];
  const float* w2 = (const float*)d_in[3];
  const float* b2 = (const float*)d_in[4];
  const float* wf = (const float*)d_in[5];
  const float* bf = (const float*)d_in[6];
  float* out = (float*)d_out;

  float* h1p = (float*)d_ws;                            // 8*256*4128  (33.8 MB)
  float* hp  = h1p + (size_t)BB * HH * H1P;             // 8*512*4112  (67.4 MB)
  float* xt  = hp  + (size_t)BB * CC2 * HPP;            // 8*32*4112   ( 4.2 MB)
  float* wp1 = xt  + (size_t)BB * CIN * XTP;            // 256*384     ( 0.4 MB)
  float* wp3 = wp1 + (size_t)HH * K1ROW;                // 32*6144     ( 0.8 MB)

  pack_w1_k      <<<(HH * K1ROW) / 256, 256, 0, stream>>>(w1, wp1);
  pack_w3_k      <<<(32 * K3ROW) / 256, 256, 0, stream>>>(wf, wp3);
  transpose_pad_k<<<(BB * CIN * XTP) / 256, 256, 0, stream>>>(x, xt);
  zero_h1_halo_k <<<(BB * HH * 19) / 256, 256, 0, stream>>>(h1p);

  conv1_k<<<(8  * BB * 64) / 8, 256, 0, stream>>>(xt, wp1, b1, h1p);
  conv2_k<<<(16 * BB * 64) / 8, 256, 0, stream>>>(h1p, w2, b2, hp);
  fix_hp_k<<<(BB * CC2 * 16) / 256, 256, 0, stream>>>(hp);
  conv3_k<<<(BB * 64) / 8, 256, 0, stream>>>(hp, wp3, bf, out);
}